// ConditionalFeedForward_69449621176928
// MI455X (gfx1250) — compile-verified
//
#include <hip/hip_runtime.h>
#include <hip/hip_bf16.h>
#include <math.h>
#include <stdint.h>

// Problem constants (from reference): M=2048, D=1024, I=2048, E=8, TOPK=2
#define M_TOK   2048
#define DMODEL  1024
#define IDIM    2048
#define NEXP    8
#define TOPK    2
#define NASSIGN (M_TOK * TOPK)   // 4096 routed (token, slot) assignments
#define PADK    68               // 68 floats = 272 B row stride: 16B-aligned, conflict-free

typedef __attribute__((ext_vector_type(2))) float v2f;
typedef __attribute__((ext_vector_type(8))) float v8f;

// ---- workspace layout ----
// [0,32)                : per-expert counters (8 x i32)
// [64, 64+8*4096*4)     : per-expert assignment lists (128 KB)
// [256KB, 256KB+32MB)   : fp32 activation buffer, NASSIGN x IDIM
#define WS_COUNTS 0
#define WS_LISTS  64
#define WS_ACT    (256 * 1024)

__device__ __forceinline__ v8f wmma_f32_4(v2f a, v2f b, v8f c) {
  // V_WMMA_F32_16X16X4_F32 : D(16x16 f32) = A(16x4 f32) x B(4x16 f32) + C
  return __builtin_amdgcn_wmma_f32_16x16x4_f32(
      false, a, false, b, (short)0, c, false, false);
}

// GLOBAL_LOAD_ASYNC_TO_LDS_B128 (GVS mode): per-lane 32-bit byte offset +
// uniform 64-bit SGPR base; destination is a per-lane LDS byte address.
// Tracked with ASYNCcnt (CDNA5 ISA).
__device__ __forceinline__ void async_ld_b128(uint32_t lds_addr, uint32_t voff,
                                              const void* sbase) {
  asm volatile("global_load_async_to_lds_b128 %0, %1, %2 offset:0"
               :: "v"(lds_addr), "v"(voff), "s"(sbase)
               : "memory");
}
#define WAIT_ASYNC(n) asm volatile("s_wait_asynccnt " #n ::: "memory")

__device__ __forceinline__ uint32_t lds_addr_of(const void* p) {
  return (uint32_t)(uintptr_t)p;  // LDS aperture: addr[31:0] is the LDS offset
}

// -------------------------------------------------------------------------
// Kernel 1: routing. Compact assignment ids (m*TOPK+k) into per-expert lists.
// List order is nondeterministic (atomics) but the final output is not:
// every act/out row is keyed by its assignment id and computed exactly once.
// -------------------------------------------------------------------------
__global__ __launch_bounds__(256) void route_kernel(
    const int* __restrict__ idx, int* __restrict__ counts,
    int* __restrict__ lists) {
  int a = blockIdx.x * blockDim.x + threadIdx.x;
  if (a < NASSIGN) {
    int e = idx[a];
    int pos = atomicAdd(&counts[e], 1);
    lists[e * NASSIGN + pos] = a;
  }
}

// -------------------------------------------------------------------------
// Kernel 2: h = x . w13^T (w1 and w3 halves), SwiGLU, scatter to act buffer.
// Double-buffered async global->LDS pipeline: 12 async B128 per wave per
// K-chunk (4 for gathered x rows, 8 for w1/w3 rows).
// -------------------------------------------------------------------------
__global__ __launch_bounds__(256) void moe_gemm1_swiglu(
    const float* __restrict__ x, const float* __restrict__ w13,
    const int* __restrict__ counts, const int* __restrict__ lists,
    float* __restrict__ act) {
  const int e  = blockIdx.z;
  const int mt = blockIdx.x;
  const int nt = blockIdx.y;
  const int cnt = counts[e];
  if (mt * 64 >= cnt) return;          // uniform exit: EXEC stays all-ones
  const int n0 = nt * 64;

  __shared__ float Xs[2][64][PADK];
  __shared__ float Ws[2][128][PADK];
  __shared__ int   rowAid[64];

  const int tid  = threadIdx.x;
  const int lane = tid & 31;
  const int wid  = tid >> 5;           // 0..7
  const int mi   = wid & 3;            // token 16-block within tile
  const int ng   = wid >> 2;           // 0..1 : which 32-col half
  const int lr   = lane & 15;
  const int hi   = (lane >> 4) & 1;    // half-wave: K+2 / M+8 selector

  if (tid < 64) {
    int slot = mt * 64 + tid;
    int cs = slot < cnt ? slot : (cnt - 1);   // clamped rows computed, never stored
    rowAid[tid] = lists[e * NASSIGN + cs];
  }

  // fixed per-thread async-load coordinates
  int xrow[4], xkq[4], xmrow[4];
#pragma unroll
  for (int i = 0; i < 4; ++i) {
    int lin = tid + i * 256; xrow[i] = lin >> 4; xkq[i] = (lin & 15) * 4;
  }
  int wrowl[8], wkq[8], wrowg[8];
#pragma unroll
  for (int i = 0; i < 8; ++i) {
    int lin = tid + i * 256; int row = lin >> 4;
    wrowl[i] = row; wkq[i] = (lin & 15) * 4;
    wrowg[i] = (row < 64) ? (n0 + row) : (IDIM + n0 + (row - 64));
  }
  __syncthreads();                      // rowAid visible
#pragma unroll
  for (int i = 0; i < 4; ++i) xmrow[i] = rowAid[xrow[i]] / TOPK;

  const float* w13e = w13 + (size_t)e * (2 * IDIM) * DMODEL;
  const uint32_t xsb = lds_addr_of(&Xs[0][0][0]);
  const uint32_t wsb = lds_addr_of(&Ws[0][0][0]);

  auto prefetch = [&](int b, int k0) {
#pragma unroll
    for (int i = 0; i < 4; ++i)
      async_ld_b128(xsb + (uint32_t)(((b * 64 + xrow[i]) * PADK + xkq[i]) * 4),
                    (uint32_t)((xmrow[i] * DMODEL + k0 + xkq[i]) * 4), x);
#pragma unroll
    for (int i = 0; i < 8; ++i)
      async_ld_b128(wsb + (uint32_t)(((b * 128 + wrowl[i]) * PADK + wkq[i]) * 4),
                    (uint32_t)((wrowg[i] * DMODEL + k0 + wkq[i]) * 4), w13e);
  };

  v8f c1[2] = {};
  v8f c3[2] = {};

  prefetch(0, 0);
  for (int kc = 0; kc < DMODEL / 64; ++kc) {
    if (kc + 1 < DMODEL / 64) {
      prefetch((kc + 1) & 1, (kc + 1) * 64);
      WAIT_ASYNC(12);                  // chunk kc's 12 in-order ops have landed
    } else {
      WAIT_ASYNC(0);
    }
    __syncthreads();                   // all waves' LDS writes visible
    const int b = kc & 1;
#pragma unroll 4
    for (int k4 = 0; k4 < 16; ++k4) {
      const int kk = k4 * 4 + hi * 2;
      v2f a;
      a.x = Xs[b][mi * 16 + lr][kk];
      a.y = Xs[b][mi * 16 + lr][kk + 1];
#pragma unroll
      for (int t = 0; t < 2; ++t) {
        const int nc = ng * 32 + t * 16 + lr;
        v2f b1, b3;
        b1.x = Ws[b][nc][kk];      b1.y = Ws[b][nc][kk + 1];
        b3.x = Ws[b][64 + nc][kk]; b3.y = Ws[b][64 + nc][kk + 1];
        c1[t] = wmma_f32_4(a, b1, c1[t]);
        c3[t] = wmma_f32_4(a, b3, c3[t]);
      }
    }
    __syncthreads();                   // reads done before buffer b is recycled
  }

  // SwiGLU in registers, scatter to act buffer
#pragma unroll
  for (int t = 0; t < 2; ++t) {
#pragma unroll
    for (int r = 0; r < 8; ++r) {
      float h1 = c1[t][r];
      float h3 = c3[t][r];
      float av = h1 * __builtin_amdgcn_rcpf(1.f + __expf(-h1)) * h3;
      int mrow = mi * 16 + r + hi * 8;
      int slot = mt * 64 + mrow;
      if (slot < cnt) {
        int aid = rowAid[mrow];
        act[(size_t)aid * IDIM + n0 + ng * 32 + t * 16 + lr] = av;
      }
    }
  }
}

// -------------------------------------------------------------------------
// Kernel 3: out = act . w2^T, scattered to out[aid*D + d].
// Same double-buffered async pipeline, 8 async B128 per wave per K-chunk.
// -------------------------------------------------------------------------
__global__ __launch_bounds__(256) void moe_gemm2(
    const float* __restrict__ act, const float* __restrict__ w2,
    const int* __restrict__ counts, const int* __restrict__ lists,
    float* __restrict__ out) {
  const int e  = blockIdx.z;
  const int mt = blockIdx.x;
  const int dt = blockIdx.y;
  const int cnt = counts[e];
  if (mt * 64 >= cnt) return;
  const int d0 = dt * 64;

  __shared__ float As[2][64][PADK];
  __shared__ float Bs[2][64][PADK];
  __shared__ int   rowAid[64];

  const int tid  = threadIdx.x;
  const int lane = tid & 31;
  const int wid  = tid >> 5;
  const int mi   = wid & 3;
  const int nh   = wid >> 2;           // 0..1
  const int lr   = lane & 15;
  const int hi   = (lane >> 4) & 1;

  if (tid < 64) {
    int slot = mt * 64 + tid;
    int cs = slot < cnt ? slot : (cnt - 1);
    rowAid[tid] = lists[e * NASSIGN + cs];
  }

  int arow[4], akq[4], aaid[4], brow[4];
#pragma unroll
  for (int i = 0; i < 4; ++i) {
    int lin = tid + i * 256; arow[i] = lin >> 4; akq[i] = (lin & 15) * 4;
    brow[i] = d0 + arow[i];
  }
  __syncthreads();
#pragma unroll
  for (int i = 0; i < 4; ++i) aaid[i] = rowAid[arow[i]];

  const float* w2e = w2 + (size_t)e * DMODEL * IDIM;
  const uint32_t asb = lds_addr_of(&As[0][0][0]);
  const uint32_t bsb = lds_addr_of(&Bs[0][0][0]);

  auto prefetch = [&](int b, int k0) {
#pragma unroll
    for (int i = 0; i < 4; ++i)
      async_ld_b128(asb + (uint32_t)(((b * 64 + arow[i]) * PADK + akq[i]) * 4),
                    (uint32_t)((aaid[i] * IDIM + k0 + akq[i]) * 4), act);
#pragma unroll
    for (int i = 0; i < 4; ++i)
      async_ld_b128(bsb + (uint32_t)(((b * 64 + arow[i]) * PADK + akq[i]) * 4),
                    (uint32_t)((brow[i] * IDIM + k0 + akq[i]) * 4), w2e);
  };

  v8f c[2] = {};

  prefetch(0, 0);
  for (int kc = 0; kc < IDIM / 64; ++kc) {
    if (kc + 1 < IDIM / 64) {
      prefetch((kc + 1) & 1, (kc + 1) * 64);
      WAIT_ASYNC(8);
    } else {
      WAIT_ASYNC(0);
    }
    __syncthreads();
    const int b = kc & 1;
#pragma unroll 4
    for (int k4 = 0; k4 < 16; ++k4) {
      const int kk = k4 * 4 + hi * 2;
      v2f a;
      a.x = As[b][mi * 16 + lr][kk];
      a.y = As[b][mi * 16 + lr][kk + 1];
#pragma unroll
      for (int t = 0; t < 2; ++t) {
        const int nc = nh * 32 + t * 16 + lr;
        v2f bb;
        bb.x = Bs[b][nc][kk]; bb.y = Bs[b][nc][kk + 1];
        c[t] = wmma_f32_4(a, bb, c[t]);
      }
    }
    __syncthreads();
  }

#pragma unroll
  for (int t = 0; t < 2; ++t) {
#pragma unroll
    for (int r = 0; r < 8; ++r) {
      int mrow = mi * 16 + r + hi * 8;
      int slot = mt * 64 + mrow;
      if (slot < cnt) {
        int aid = rowAid[mrow];
        out[(size_t)aid * DMODEL + d0 + nh * 32 + t * 16 + lr] = c[t][r];
      }
    }
  }
}

extern "C" void kernel_launch(void* const* d_in, const int* in_sizes, int n_in,
                              void* d_out, int out_size, void* d_ws, size_t ws_size,
                              hipStream_t stream) {
  (void)in_sizes; (void)n_in; (void)out_size; (void)ws_size;
  const float* x   = (const float*)d_in[0];       // (M, D)
  const float* w13 = (const float*)d_in[1];       // (E, 2I, D)
  const float* w2  = (const float*)d_in[2];       // (E, D, I)
  const int*   idx = (const int*)d_in[3];         // (M, TOPK)
  float* out = (float*)d_out;                     // (M, TOPK, D) fp32

  char* ws = (char*)d_ws;
  int*   counts = (int*)(ws + WS_COUNTS);
  int*   lists  = (int*)(ws + WS_LISTS);
  float* act    = (float*)(ws + WS_ACT);

  hipMemsetAsync(counts, 0, NEXP * sizeof(int), stream);
  route_kernel<<<dim3((NASSIGN + 255) / 256), dim3(256), 0, stream>>>(idx, counts, lists);
  // worst case: all 4096 assignments on one expert -> 64 m-tiles; dead tiles exit early
  moe_gemm1_swiglu<<<dim3(64, IDIM / 64, NEXP), dim3(256), 0, stream>>>(x, w13, counts, lists, act);
  moe_gemm2<<<dim3(64, DMODEL / 64, NEXP), dim3(256), 0, stream>>>(act, w2, counts, lists, out);
}